// FixedRadiusNearNeighbors_46170898432073
// MI455X (gfx1250) — compile-verified
//
#include <hip/hip_runtime.h>

// Problem constants (fixed in the reference file).
#define G_  40
#define P_  16
#define K_  32
#define B_  4
#define N_  65536
#define S_  4096

#define WAVES_PER_BLOCK 8   // 256 threads = 8 wave32 waves, one centroid per wave

// Wave-local ordering fence: prevents compiler reordering / store-forwarding of
// LDS ops across phases. Hardware keeps same-wave LDS ops in order, so no
// cross-wave barrier is needed (each wave owns its private LDS row).
__device__ __forceinline__ void wave_lds_fence() {
    __builtin_amdgcn_wave_barrier();
    asm volatile("" ::: "memory");
}

__global__ __launch_bounds__(32 * WAVES_PER_BLOCK)
void frnn_gather_kernel(const float* __restrict__ pos,
                        const int*   __restrict__ centroids,
                        const int*   __restrict__ voxel_points,
                        const int*   __restrict__ voxel_counts,
                        float*       __restrict__ out)
{
    // LDS staging: one K-slot row per wave; layout identical to global output,
    // so the final store is fully coalesced. No row is shared across waves.
    __shared__ float s_out[WAVES_PER_BLOCK * K_];

    const int lane = threadIdx.x & 31;
    const int wave = threadIdx.x >> 5;
    const int cidx = blockIdx.x * WAVES_PER_BLOCK + wave;   // 0 .. B*S-1
    const int b    = cidx >> 12;                            // / S_ (4096)

    // Centroid index + its position (uniform across the wave; L2-resident).
    const int c = centroids[cidx];
    const float* pp = pos + ((size_t)b * N_ + (size_t)c) * 3;
    const float px = pp[0], py = pp[1], pz = pp[2];
    const int vx = (int)(px * (G_ - 1));
    const int vy = (int)(py * (G_ - 1));
    const int vz = (int)(pz * (G_ - 1));

    // Phase 1: default-fill this wave's output row with the centroid index.
    s_out[wave * K_ + lane] = (float)c;
    wave_lds_fence();

    // Phase 2: lanes 0..26 each own one neighbor voxel, in the reference's
    // flat (dx outer, dy, dz inner) order. All 27 count gathers issue at once.
    int    cnt   = 0;
    size_t vbase = 0;
    if (lane < 27) {
        const int dz = (lane % 3) - 1;
        const int dy = ((lane / 3) % 3) - 1;
        const int dx = (lane / 9) - 1;
        const int nx = vx + dx, ny = vy + dy, nz = vz + dz;
        if ((unsigned)nx < G_ && (unsigned)ny < G_ && (unsigned)nz < G_) {
            const size_t vox = (((size_t)b * G_ + nx) * G_ + ny) * G_ + nz;
            cnt = voxel_counts[vox];             // mask>0 is implied by cnt>0
            vbase = vox * P_;
            if (cnt > 0) {
                // gfx1250 global_prefetch_b8: pull the 64B candidate row now,
                // hiding its latency behind the shuffle scan below.
                __builtin_prefetch(voxel_points + vbase, 0, 1);
            }
        }
    }

    // Exclusive prefix sum of counts across the wave (5-step shuffle scan).
    int incl = cnt;
    #pragma unroll
    for (int d = 1; d < 32; d <<= 1) {
        const int t = __shfl_up(incl, d, 32);
        if (lane >= d) incl += t;
    }
    const int excl = incl - cnt;

    // Phase 3: scatter this voxel's candidates into the staging row, clamped
    // so total kept candidates never exceed K. Rows are 64B-aligned, so read
    // them as int4 (global_load_b128): <=4 vector loads instead of <=16 scalar.
    const int lim = K_ - excl;                    // may be <= 0
    const int m   = (cnt < lim) ? cnt : lim;
    if (m > 0) {
        const int4* __restrict__ row = (const int4*)(voxel_points + vbase);
        float* dst = &s_out[wave * K_ + excl];
        for (int q = 0; q < m; q += 4) {
            const int4 v4 = row[q >> 2];
            dst[q] = (float)v4.x;                 // q < m guaranteed here
            if (q + 1 < m) dst[q + 1] = (float)v4.y;
            if (q + 2 < m) dst[q + 2] = (float)v4.z;
            if (q + 3 < m) dst[q + 3] = (float)v4.w;
        }
    }
    wave_lds_fence();

    // Phase 4: fully coalesced store — this wave's 32 outputs are contiguous
    // in global memory and in LDS (row == threadIdx.x slice).
    out[(size_t)blockIdx.x * (32 * WAVES_PER_BLOCK) + threadIdx.x] = s_out[threadIdx.x];
}

extern "C" void kernel_launch(void* const* d_in, const int* in_sizes, int n_in,
                              void* d_out, int out_size, void* d_ws, size_t ws_size,
                              hipStream_t stream)
{
    (void)in_sizes; (void)n_in; (void)out_size; (void)d_ws; (void)ws_size;

    // setup_inputs() order:
    //   0: pos (B,N,3) f32        1: centroids (B,S) i32
    //   2: voxel_points (B,G,G,G,P) i32   3: voxel_counts (B,G,G,G) i32
    //   4: neighbour_voxel_list (unused: equals coords+offsets by construction)
    //   5: mask (unused: equals counts>0)  6: voxel_size scalar (G=40, fixed)
    const float* pos          = (const float*)d_in[0];
    const int*   centroids    = (const int*)  d_in[1];
    const int*   voxel_points = (const int*)  d_in[2];
    const int*   voxel_counts = (const int*)  d_in[3];
    float*       out          = (float*)      d_out;

    const int total_centroids = B_ * S_;                  // 16384
    dim3 grid(total_centroids / WAVES_PER_BLOCK);         // 2048 blocks
    dim3 block(32 * WAVES_PER_BLOCK);                     // 256 threads (8 waves)
    frnn_gather_kernel<<<grid, block, 0, stream>>>(pos, centroids, voxel_points,
                                                   voxel_counts, out);
}